// NavierStokesGradientLayer_50749333569980
// MI455X (gfx1250) — compile-verified
//
#include <hip/hip_runtime.h>

// PINN Navier-Stokes derivative stack on gfx1250 (CDNA5).
// Full 3rd-order forward-mode jet (10 channels) propagated through the MLP
// with V_WMMA_F32_16X16X4_F32. Points ride lanes; the C/D register layout of
// layer l is reused verbatim as the B layout of layer l+1 by absorbing the
// K-slot unit permutation u(kb,j) into the W^T (A-matrix) load addressing.
// => no LDS, no cross-lane shuffles between layers.
//
// Round-2 change: all weight/bias addresses are formed as
//   (single lane-dependent base VGPR) + (compile-time constant)
// so the addressing collapses to one base register + 24-bit immediates,
// eliminating the address-spill scratch traffic seen in round 1.

typedef float v2f __attribute__((ext_vector_type(2)));
typedef float v8f __attribute__((ext_vector_type(8)));

struct PinnParams {
  const float* xy;     // (N,2)
  const float* W[7];   // dims: 2x64, 5x(64x64), 64x2 (row major: W[in][out])
  const float* b[7];
  float* out;          // 14 arrays of N, concatenated
  int N;
};

static __device__ __forceinline__ v8f wmma4(v2f a, v2f b, v8f c) {
  // D(16x16 f32) = A(16x4 f32) x B(4x16 f32) + C
  return __builtin_amdgcn_wmma_f32_16x16x4_f32(false, a, false, b,
                                               (short)0, c, false, false);
}

static __device__ __forceinline__ float fast_tanh(float x) {
#if __has_builtin(__builtin_amdgcn_tanh_f32)
  return __builtin_amdgcn_tanh_f32(x);
#else
  // tanh(x) = 1 - 2/(exp(2x)+1),  exp(2x) = exp2(2*log2(e)*x)
  float e = __builtin_amdgcn_exp2f(x * 2.8853900817779268f);
  return 1.0f - 2.0f * __builtin_amdgcn_rcpf(e + 1.0f);
#endif
}

__global__ void __launch_bounds__(256, 1)
pinn_jet_kernel(PinnParams p) {
  const int lane  = threadIdx.x & 31;
  const int wave  = threadIdx.x >> 5;
  const int tile  = blockIdx.x * 8 + wave;      // 16 points per wave-tile
  if (tile * 16 >= p.N) return;                 // uniform per wave

  const int m     = lane & 15;                  // point-in-tile / A row
  const int uoff8 = (lane & 16) ? 8 : 0;        // hi-half-of-wave unit offset
  const int pt0   = tile * 16;

  // Single lane-dependent address bases; everything else is an immediate.
  const int abase = uoff8 * 64 + m;             // hidden-layer W^T loads
  const int ubase = uoff8;                      // layer-0 / bias loads

  // Both lane halves carry the same 16 points (B/D layout: point = lane%16).
  v2f xy2 = *(const v2f*)(p.xy + (size_t)(pt0 + m) * 2);
  const float x = xy2[0], y = xy2[1];

  // h[channel][nt] : 16x16 C/D-layout tile, units 16nt..16nt+15, pts 0..15.
  // channels: 0=val 1=dx 2=dy 3=dxx 4=dxy 5=dyy 6=dxxx 7=dxxy 8=dxyy 9=dyyy
  v8f h[10][4];

  // ---------------- layer 0 (2 -> 64, tanh): analytic jet -----------------
  {
    const float* W0 = p.W[0];
    const float* B0 = p.b[0];
#pragma unroll
    for (int nt = 0; nt < 4; ++nt) {
#pragma unroll
      for (int r = 0; r < 8; ++r) {
        float a  = W0[ubase + (nt * 16 + r)];        // dz/dx (const per unit)
        float bb = W0[ubase + (64 + nt * 16 + r)];   // dz/dy
        float bs = B0[ubase + (nt * 16 + r)];
        float z  = fmaf(a, x, fmaf(bb, y, bs));
        float s    = fast_tanh(z);
        float sp   = fmaf(-s, s, 1.0f);              // s'
        float spp  = -2.0f * s * sp;                 // s''
        float sppp = -2.0f * fmaf(sp, sp, s * spp);  // s'''
        h[0][nt][r] = s;
        h[1][nt][r] = sp * a;
        h[2][nt][r] = sp * bb;
        h[3][nt][r] = spp * a * a;
        h[4][nt][r] = spp * a * bb;
        h[5][nt][r] = spp * bb * bb;
        h[6][nt][r] = sppp * a * a * a;
        h[7][nt][r] = sppp * a * a * bb;
        h[8][nt][r] = sppp * a * bb * bb;
        h[9][nt][r] = sppp * bb * bb * bb;
      }
    }
  }

  // ------------- layers 1..5 (64 -> 64, tanh): WMMA + jet -----------------
  const float* Wl = p.W[1];
  const float* Bl = p.b[1];
  for (int l = 1; l <= 5; ++l) {
    // A = W^T tiles with permuted K-slot unit order u(kb,j):
    //   lane-dependent base: abase = uoff8*64 + m; all else immediate.
    v2f A[16][4];
#pragma unroll
    for (int kb = 0; kb < 16; ++kb) {
      const int kc = ((kb >> 2) * 16 + (kb & 3) * 2) * 64;   // constant
#pragma unroll
      for (int nt = 0; nt < 4; ++nt) {
        v2f a;
        a[0] = Wl[abase + (kc + nt * 16)];
        a[1] = Wl[abase + (kc + 64 + nt * 16)];
        A[kb][nt] = a;
      }
    }
    // bias in C/D layout (channel 0 only) -- transient, dead after c=0
    v8f bv[4];
#pragma unroll
    for (int nt = 0; nt < 4; ++nt)
#pragma unroll
      for (int r = 0; r < 8; ++r)
        bv[nt][r] = Bl[ubase + (nt * 16 + r)];

    // GEMM: c-outer so h[c] dies exactly as acc[c] is born (reg overlap).
    v8f acc[10][4];
#pragma unroll
    for (int c = 0; c < 10; ++c) {
#pragma unroll
      for (int nt = 0; nt < 4; ++nt) {
        v8f d = (c == 0) ? bv[nt] : (v8f){0, 0, 0, 0, 0, 0, 0, 0};
#pragma unroll
        for (int kb = 0; kb < 16; ++kb) {
          v2f bb;                       // B tile = slice of prev D, verbatim
          bb[0] = h[c][kb >> 2][(kb & 3) * 2 + 0];
          bb[1] = h[c][kb >> 2][(kb & 3) * 2 + 1];
          d = wmma4(A[kb][nt], bb, d);
        }
        acc[c][nt] = d;
      }
    }

    // tanh jet (elementwise in C/D layout), writes back into h
#pragma unroll
    for (int nt = 0; nt < 4; ++nt) {
#pragma unroll
      for (int r = 0; r < 8; ++r) {
        float z1 = acc[1][nt][r], z2 = acc[2][nt][r];
        float z3 = acc[3][nt][r], z4 = acc[4][nt][r], z5 = acc[5][nt][r];
        float z6 = acc[6][nt][r], z7 = acc[7][nt][r], z8 = acc[8][nt][r];
        float z9 = acc[9][nt][r];
        float s    = fast_tanh(acc[0][nt][r]);
        float sp   = fmaf(-s, s, 1.0f);
        float spp  = -2.0f * s * sp;
        float sppp = -2.0f * fmaf(sp, sp, s * spp);
        h[0][nt][r] = s;
        h[1][nt][r] = sp * z1;
        h[2][nt][r] = sp * z2;
        h[3][nt][r] = fmaf(spp, z1 * z1, sp * z3);
        h[4][nt][r] = fmaf(spp, z1 * z2, sp * z4);
        h[5][nt][r] = fmaf(spp, z2 * z2, sp * z5);
        h[6][nt][r] = fmaf(sppp, z1 * z1 * z1,
                       fmaf(3.0f * spp, z1 * z3, sp * z6));
        h[7][nt][r] = fmaf(sppp, z1 * z1 * z2,
                       fmaf(spp, fmaf(2.0f, z1 * z4, z3 * z2), sp * z7));
        h[8][nt][r] = fmaf(sppp, z1 * z2 * z2,
                       fmaf(spp, fmaf(2.0f, z2 * z4, z5 * z1), sp * z8));
        h[9][nt][r] = fmaf(sppp, z2 * z2 * z2,
                       fmaf(3.0f * spp, z2 * z5, sp * z9));
      }
    }

    // advance weight pointers with compile-time kernarg indices
    Wl = (l == 1) ? p.W[2] : (l == 2) ? p.W[3] : (l == 3) ? p.W[4] : p.W[5];
    Bl = (l == 1) ? p.b[2] : (l == 2) ? p.b[3] : (l == 3) ? p.b[4] : p.b[5];
  }

  // --------------- final layer (64 -> 2, linear): WMMA --------------------
  const float* W6 = p.W[6];
  const float* B6 = p.b[6];
  const int fbase = uoff8 * 2 + ((m < 2) ? m : 0);  // single lane-dep base
  v2f A7[16];
#pragma unroll
  for (int kb = 0; kb < 16; ++kb) {
    const int kc = ((kb >> 2) * 16 + (kb & 3) * 2) * 2;     // constant
    float w0 = W6[fbase + kc];
    float w1 = W6[fbase + (kc + 2)];
    v2f a;
    a[0] = (m < 2) ? w0 : 0.0f;          // pad rows M>=2 with zeros
    a[1] = (m < 2) ? w1 : 0.0f;
    A7[kb] = a;
  }
  float psi[10], pp[3];
#pragma unroll
  for (int c = 0; c < 10; ++c) {
    v8f d = {0, 0, 0, 0, 0, 0, 0, 0};
#pragma unroll
    for (int kb = 0; kb < 16; ++kb) {
      v2f bb;
      bb[0] = h[c][kb >> 2][(kb & 3) * 2 + 0];
      bb[1] = h[c][kb >> 2][(kb & 3) * 2 + 1];
      d = wmma4(A7[kb], bb, d);
    }
    psi[c] = d[0];                 // M=0 row, lanes 0..15 = points
    if (c < 3) pp[c] = d[1];       // M=1 row
  }
  psi[0] += B6[0];
  pp[0]  += B6[1];

  // ------------------------------ stores -----------------------------------
  if (lane < 16) {
    float* out = p.out;
    const int N = p.N;
    const int idx = pt0 + m;
    out[0  * N + idx] =  psi[0];   // psi
    out[1  * N + idx] =  pp[0];    // p
    out[2  * N + idx] =  pp[1];    // dp/dx
    out[3  * N + idx] =  pp[2];    // dp/dy
    out[4  * N + idx] =  psi[2];   // u = psi_y
    out[5  * N + idx] =  psi[4];   // du/dx = psi_xy
    out[6  * N + idx] =  psi[5];   // du/dy = psi_yy
    out[7  * N + idx] =  psi[7];   // du/dx2 = psi_xxy
    out[8  * N + idx] =  psi[9];   // du/dy2 = psi_yyy
    out[9  * N + idx] = -psi[1];   // v = -psi_x
    out[10 * N + idx] = -psi[3];   // dv/dx = -psi_xx
    out[11 * N + idx] = -psi[4];   // dv/dy = -psi_xy
    out[12 * N + idx] = -psi[6];   // dv/dx2 = -psi_xxx
    out[13 * N + idx] = -psi[8];   // dv/dy2 = -psi_xyy
  }
}

extern "C" void kernel_launch(void* const* d_in, const int* in_sizes, int n_in,
                              void* d_out, int out_size, void* d_ws,
                              size_t ws_size, hipStream_t stream) {
  (void)n_in; (void)out_size; (void)d_ws; (void)ws_size;
  PinnParams p;
  p.xy = (const float*)d_in[0];
  for (int i = 0; i < 7; ++i) p.W[i] = (const float*)d_in[1 + i];
  for (int i = 0; i < 7; ++i) p.b[i] = (const float*)d_in[8 + i];
  p.out = (float*)d_out;
  p.N = in_sizes[0] / 2;                       // xy is (N,2)

  const int tiles = p.N / 16;                  // 131072/16 = 8192
  const int wavesPerBlock = 8;                 // 256 threads, wave32
  const int blocks = (tiles + wavesPerBlock - 1) / wavesPerBlock;
  pinn_jet_kernel<<<blocks, 32 * wavesPerBlock, 0, stream>>>(p);
}